// WaveletFilterNet_40269613367557
// MI455X (gfx1250) — compile-verified
//
#include <hip/hip_runtime.h>
#include <cstdint>
#include <cstddef>

// WaveletFilterNet: fused Haar forward -> per-band filter -> Haar inverse.
// Memory-bound elementwise 2x2-block transform. CDNA5 path: double-buffered
// global_load_async_to_lds_b128 pipeline over the 16-image batch loop.

typedef float v4f __attribute__((ext_vector_type(4)));

#define IMG_H 2048
#define IMG_W 2048
#define H2 1024
#define W2 1024
#define NBATCH 16

__device__ __forceinline__ void async_copy_b128(uint32_t lds_byte_addr,
                                                const float* gaddr) {
    // GLOBAL_LOAD_ASYNC_TO_LDS_B128: each lane copies 16B from global vaddr to LDS address vdst.
    // Tracked by ASYNCcnt, not LOADcnt.
    asm volatile("global_load_async_to_lds_b128 %0, %1, off"
                 :: "v"(lds_byte_addr), "v"(gaddr)
                 : "memory");
}

__global__ __launch_bounds__(256) void
haar_filter_kernel(const float* __restrict__ x,
                   const float* __restrict__ filt,
                   float* __restrict__ out) {
    // LDS: [2 pipeline buffers][2 rows][1024 floats] = 16 KiB
    __shared__ __align__(16) float smem[2 * 2 * 1024];

    const int t  = threadIdx.x;            // 0..255
    const int i  = blockIdx.y;             // block-row in h2 (0..1023)
    const int j2 = blockIdx.x * 512 + 2 * t; // this thread's first 2x2-block col
    const int xc = 2 * j2;                 // x column base (4 consecutive floats)
    const int r0 = 2 * i;                  // top x row

    // Per-thread filter values, loaded once, reused across all 16 batches.
    const size_t frow = (size_t)i * W2 + (size_t)j2;
    const float f00 = filt[0 * (size_t)H2 * W2 + frow];
    const float f01 = filt[0 * (size_t)H2 * W2 + frow + 1];
    const float f10 = filt[1 * (size_t)H2 * W2 + frow];
    const float f11 = filt[1 * (size_t)H2 * W2 + frow + 1];
    const float f20 = filt[2 * (size_t)H2 * W2 + frow];
    const float f21 = filt[2 * (size_t)H2 * W2 + frow + 1];

    // LDS byte offset of smem (flat LDS addresses: low 32 bits = LDS offset).
    const uint32_t lds_base = (uint32_t)(uintptr_t)&smem[0];

    const size_t img = (size_t)IMG_H * IMG_W;
    const size_t rowbase0 = (size_t)r0 * IMG_W + (size_t)xc;       // top row
    const size_t rowbase1 = (size_t)(r0 + 1) * IMG_W + (size_t)xc; // bottom row

    // Prologue: kick off batch 0 into buffer 0.
    async_copy_b128(lds_base + (uint32_t)(t * 16),        x + rowbase0);
    async_copy_b128(lds_base + 4096 + (uint32_t)(t * 16), x + rowbase1);

    for (int b = 0; b < NBATCH; ++b) {
        const uint32_t buf = (uint32_t)(b & 1);

        if (b + 1 < NBATCH) {
            // Make sure our previous-iteration LDS reads have retired before
            // the async engine may overwrite that buffer.
            asm volatile("s_wait_dscnt 0x0" ::: "memory");
            const uint32_t nbuf = buf ^ 1u;
            const size_t nbase = (size_t)(b + 1) * img;
            async_copy_b128(lds_base + nbuf * 8192 + (uint32_t)(t * 16),
                            x + nbase + rowbase0);
            async_copy_b128(lds_base + nbuf * 8192 + 4096 + (uint32_t)(t * 16),
                            x + nbase + rowbase1);
            // Allow the 2 just-issued ops to remain in flight; guarantees the
            // 2 older ops (this batch's data) have landed in LDS.
            asm volatile("s_wait_asynccnt 0x2" ::: "memory");
        } else {
            asm volatile("s_wait_asynccnt 0x0" ::: "memory");
        }

        // Read this thread's staged 2x4 pixel patch from LDS.
        const v4f top = *reinterpret_cast<const v4f*>(&smem[buf * 2048 + t * 4]);
        const v4f bot = *reinterpret_cast<const v4f*>(&smem[buf * 2048 + 1024 + t * 4]);

        // ---- block k = 0 (columns j2) ----
        float a0 = top.x, b0 = top.y, c0 = bot.x, d0 = bot.y;
        float LL = 0.5f * (a0 + b0 + c0 + d0);
        float LH = 0.5f * (a0 - b0 + c0 - d0);
        float HL = 0.5f * (a0 + b0 - c0 - d0);
        float HH = 0.5f * (a0 - b0 - c0 + d0);
        float u = HH * f00, v = HL * f10, w = LH * f20;
        float oa0 = 0.5f * (LL + u + v + w);
        float ob0 = 0.5f * (LL - u + v - w);
        float oc0 = 0.5f * (LL + u - v - w);
        float od0 = 0.5f * (LL - u - v + w);

        // ---- block k = 1 (columns j2+1) ----
        a0 = top.z; b0 = top.w; c0 = bot.z; d0 = bot.w;
        LL = 0.5f * (a0 + b0 + c0 + d0);
        LH = 0.5f * (a0 - b0 + c0 - d0);
        HL = 0.5f * (a0 + b0 - c0 - d0);
        HH = 0.5f * (a0 - b0 - c0 + d0);
        u = HH * f01; v = HL * f11; w = LH * f21;
        const float oa1 = 0.5f * (LL + u + v + w);
        const float ob1 = 0.5f * (LL - u + v - w);
        const float oc1 = 0.5f * (LL + u - v - w);
        const float od1 = 0.5f * (LL - u - v + w);

        // Interleaved 2x2 output blocks -> one b128 store per row.
        const v4f o_top = { oa0, ob0, oa1, ob1 };
        const v4f o_bot = { oc0, od0, oc1, od1 };
        const size_t obase = (size_t)b * img;
        __builtin_nontemporal_store(o_top,
            reinterpret_cast<v4f*>(out + obase + rowbase0));
        __builtin_nontemporal_store(o_bot,
            reinterpret_cast<v4f*>(out + obase + rowbase1));
    }
}

extern "C" void kernel_launch(void* const* d_in, const int* in_sizes, int n_in,
                              void* d_out, int out_size, void* d_ws, size_t ws_size,
                              hipStream_t stream) {
    const float* x    = (const float*)d_in[0]; // (16,1,2048,2048) f32
    const float* filt = (const float*)d_in[1]; // (1,3,1024,1024) f32
    float* out = (float*)d_out;                // (16,1,2048,2048) f32

    dim3 grid(W2 / 512, H2, 1);  // (2, 1024)
    haar_filter_kernel<<<grid, 256, 0, stream>>>(x, filt, out);
}